// XDeepFM_25580825215819
// MI455X (gfx1250) — compile-verified
//
#include <hip/hip_runtime.h>

typedef float v2f __attribute__((ext_vector_type(2)));
typedef float v8f __attribute__((ext_vector_type(8)));

// Single-wave (32-thread) fused XDeepFM forward for MI455X / gfx1250.
// Both GEMVs (fc1: 128x15, fc2: 5x128) run on the fp32 WMMA pipe
// (v_wmma_f32_16x16x4_f32). All WMMA operands are fed from LDS with
// uniform, unconditional loads (weights staged + zero-padded first), so no
// exec-mask predication appears in the matrix loops.
__global__ __launch_bounds__(32) void xdeepfm_fused_kernel(
    const float* __restrict__ dense,   // (1,15)
    const float* __restrict__ sparse,  // (1,45)
    const float* __restrict__ fc1_w,   // (128,15)
    const float* __restrict__ fc1_b,   // (128,)
    const float* __restrict__ fc2_w,   // (5,128)
    const float* __restrict__ fc2_b,   // (5,)
    const float* __restrict__ fcm_w,   // (1,21)
    const float* __restrict__ fcm_b,   // (1,)
    const float* __restrict__ fcs_w,   // (1,45)
    const float* __restrict__ fcs_b,   // (1,)
    const float* __restrict__ c1w,     // (3,5,3,5,5)
    const float* __restrict__ c1b,     // (3,5)
    const float* __restrict__ c2w,
    const float* __restrict__ c2b,
    const float* __restrict__ c3w,
    const float* __restrict__ c3b,
    float* __restrict__ out)           // (1,1)
{
    __shared__ float lw1[128 * 16];  // fc1_w padded K 15 -> 16 (zero col 15)
    __shared__ float lx[16];         // dense padded to 16
    __shared__ float lw2[5 * 128];   // fc2_w staged
    __shared__ float h[128];         // fc1 activations
    __shared__ float mrg[21];        // [y_sparse, y_mlp(5), y_cin(15)]

    const int lane = threadIdx.x;    // 0..31, one wave32
    const int half = lane >> 4;      // 0: lanes 0-15, 1: lanes 16-31
    const int sub  = lane & 15;

    // ---- Stage weights/activations into LDS (uniform, unpredicated) ----
    if (lane < 16) lx[lane] = (lane < 15) ? dense[lane] : 0.0f;
#pragma unroll
    for (int j = 0; j < 60; ++j) {               // 1920 = 60 * 32
        int idx = lane + 32 * j;
        int r = idx / 15, c = idx - r * 15;
        lw1[r * 16 + c] = fc1_w[idx];
    }
#pragma unroll
    for (int r = lane; r < 128; r += 32) lw1[r * 16 + 15] = 0.0f;
#pragma unroll
    for (int j = 0; j < 20; ++j) {               // 640 = 20 * 32
        int idx = lane + 32 * j;
        lw2[idx] = fc2_w[idx];
    }
    __syncthreads();

    // ---- fc1 via V_WMMA_F32_16X16X4_F32 ----
    // A (16x4): lane sub = row M, {VGPR0,VGPR1} = K {0,1} (lanes 0-15) /
    // {2,3} (lanes 16-31). B (4x16): dense replicated across all N columns.
    const int koff = half * 2;
    v2f bm[4];
#pragma unroll
    for (int k = 0; k < 4; ++k) {
        int kb = k * 4 + koff;
        bm[k].x = lx[kb];
        bm[k].y = lx[kb + 1];
    }

#pragma unroll
    for (int mt = 0; mt < 8; ++mt) {
        const int row = mt * 16 + sub;
        v8f c = {};
#pragma unroll
        for (int k = 0; k < 4; ++k) {
            int kb = k * 4 + koff;
            v2f a;
            a.x = lw1[row * 16 + kb];
            a.y = lw1[row * 16 + kb + 1];
            c = __builtin_amdgcn_wmma_f32_16x16x4_f32(
                    false, a, false, bm[k], (short)0, c, false, false);
        }
        // B replicated across N => every lane in a half holds the same
        // column; all lanes store identical values (no exec predication).
#pragma unroll
        for (int v = 0; v < 8; ++v) {
            int m = mt * 16 + half * 8 + v;
            float val = c[v] + fc1_b[m];
            h[m] = val > 0.0f ? val : 0.0f;       // ReLU
        }
    }
    __syncthreads();

    // ---- fc2 via WMMA: M = 5 outputs (rows >=5 clamped, results unused),
    // K = 128 in 32 chunks, 4 independent accumulators to break the
    // WMMA D->C dependency chain. ----
    const int rowc = (sub < 5 ? sub : 4) * 128;
    v8f accv[4] = {};
#pragma unroll
    for (int kc = 0; kc < 32; ++kc) {
        int kb = kc * 4 + koff;
        v2f a, b;
        a.x = lw2[rowc + kb];
        a.y = lw2[rowc + kb + 1];
        b.x = h[kb];
        b.y = h[kb + 1];
        accv[kc & 3] = __builtin_amdgcn_wmma_f32_16x16x4_f32(
                false, a, false, b, (short)0, accv[kc & 3], false, false);
    }
    v8f acc2 = (accv[0] + accv[1]) + (accv[2] + accv[3]);
    if (lane == 0) {                 // lane 0 holds M=0..7 in acc2[0..7]
#pragma unroll
        for (int v = 0; v < 5; ++v) {
            float val = acc2[v] + fc2_b[v];
            mrg[1 + v] = val > 0.0f ? val : 0.0f;
        }
    }

    // ---- sparse branch: lane 5 ----
    if (lane == 5) {
        float acc = fcs_b[0];
        for (int j = 0; j < 45; ++j) acc += sparse[j] * fcs_w[j];
        mrg[0] = acc;
    }

    // ---- CIN branch (collapsed): lanes 8..22, one output each ----
    // mat[c,:,:] == s[c], s[c] = sum_r dense[r*3+c]^2;
    // y_cin[l*5+o] = sum_k ( sum_c s[c]*sum_{h,w} W[l][k][o][c][h][w] + b[l][k][o] )
    // (AvgPool(3) and the trailing *3.0 cancel exactly).
    if (lane >= 8 && lane < 23) {
        const int t = lane - 8;
        float s0 = 0.0f, s1 = 0.0f, s2 = 0.0f;
#pragma unroll
        for (int r = 0; r < 5; ++r) {
            float a0 = lx[r * 3 + 0], a1 = lx[r * 3 + 1], a2 = lx[r * 3 + 2];
            s0 += a0 * a0; s1 += a1 * a1; s2 += a2 * a2;
        }
        float s[3] = { s0, s1, s2 };
        const int l = t / 5, o = t - l * 5;
        const float* W = (l == 0) ? c1w : (l == 1) ? c2w : c3w;
        const float* B = (l == 0) ? c1b : (l == 1) ? c2b : c3b;
        float acc = 0.0f;
        for (int k = 0; k < 3; ++k) {
            for (int cc = 0; cc < 3; ++cc) {
                const float* p = W + (((k * 5 + o) * 3 + cc) * 25);
                float ws = 0.0f;
#pragma unroll
                for (int q = 0; q < 25; ++q) ws += p[q];
                acc += s[cc] * ws;
            }
            acc += B[k * 5 + o];
        }
        mrg[6 + t] = acc;
    }
    __syncthreads();

    // ---- merge: lane 0 ----
    if (lane == 0) {
        float acc = fcm_b[0];
        for (int i = 0; i < 21; ++i) acc += mrg[i] * fcm_w[i];
        out[0] = acc;
    }
}

extern "C" void kernel_launch(void* const* d_in, const int* in_sizes, int n_in,
                              void* d_out, int out_size, void* d_ws, size_t ws_size,
                              hipStream_t stream) {
    (void)in_sizes; (void)n_in; (void)d_ws; (void)ws_size; (void)out_size;
    xdeepfm_fused_kernel<<<1, 32, 0, stream>>>(
        (const float*)d_in[0],  (const float*)d_in[1],
        (const float*)d_in[2],  (const float*)d_in[3],
        (const float*)d_in[4],  (const float*)d_in[5],
        (const float*)d_in[6],  (const float*)d_in[7],
        (const float*)d_in[8],  (const float*)d_in[9],
        (const float*)d_in[10], (const float*)d_in[11],
        (const float*)d_in[12], (const float*)d_in[13],
        (const float*)d_in[14], (const float*)d_in[15],
        (float*)d_out);
}